// CBAM_6_Residual_79740362817928
// MI455X (gfx1250) — compile-verified
//
#include <hip/hip_runtime.h>
#include <math.h>

// CBAM (channel + spatial attention) with residual, B=32 C=256 H=W=56, fp32.
// Memory-bound: ~206 MB mandatory HBM traffic (x read + out write); x (102.8 MB)
// fits in the 192 MB L2 so intermediate passes over x hit cache. The shared MLP
// is done with V_WMMA_F32_16X16X4_F32 (fp32 WMMA, exact precision).

typedef __attribute__((ext_vector_type(2))) float v2f;
typedef __attribute__((ext_vector_type(8))) float v8f;

#define HW 3136      // 56*56
#define HW4 784      // HW/4
#define CCH 256
#define NTILE 13     // ceil(3136/256)

// ---------------- Kernel 1: per-(b,c) mean & max over H*W -------------------
__global__ __launch_bounds__(256) void cbam_chpool(const float* __restrict__ x,
                                                   float* __restrict__ avg,
                                                   float* __restrict__ mxp) {
    __shared__ float s_sum[256];
    __shared__ float s_max[256];
    const int bc = blockIdx.x;                       // 0 .. B*C-1
    const float4* p4 = (const float4*)(x + (size_t)bc * HW);
    float sum = 0.0f, mx = -3.402823466e38f;
    for (int i = threadIdx.x; i < HW4; i += 256) {
        float4 v = p4[i];
        sum += (v.x + v.y) + (v.z + v.w);
        mx = fmaxf(mx, fmaxf(fmaxf(v.x, v.y), fmaxf(v.z, v.w)));
    }
    s_sum[threadIdx.x] = sum;
    s_max[threadIdx.x] = mx;
    __syncthreads();
    for (int s = 128; s > 0; s >>= 1) {
        if (threadIdx.x < s) {
            s_sum[threadIdx.x] += s_sum[threadIdx.x + s];
            s_max[threadIdx.x] = fmaxf(s_max[threadIdx.x], s_max[threadIdx.x + s]);
        }
        __syncthreads();
    }
    if (threadIdx.x == 0) {
        avg[bc] = s_sum[0] * (1.0f / (float)HW);
        mxp[bc] = s_max[0];
    }
}

// ---------------- Kernel 2: shared MLP via fp32 WMMA ------------------------
// h = relu(P @ w1^T)  (P = [avg;max], 64x256 -> 64x16)
// o = h @ w2^T        (64x16 -> 64x256); ca = sigmoid(o_avg + o_max)
// 2 waves; wave w owns batch rows 16w..16w+15 for BOTH the avg block (P rows
// 16w..) and max block (P rows 32+16w..), so the final add is register-local.
__global__ __launch_bounds__(64) void cbam_mlp_wmma(const float* __restrict__ avg,
                                                    const float* __restrict__ mxp,
                                                    const float* __restrict__ w1,   // [16,256]
                                                    const float* __restrict__ w2,   // [256,16]
                                                    float* __restrict__ ca) {       // [32,256]
    __shared__ float h_lds[64][16];
    const int tid   = threadIdx.x;
    const int wv    = tid >> 5;          // wave 0..1
    const int lane  = tid & 31;
    const int lhalf = lane & 15;
    const int khi   = (lane >> 4) << 1;  // 0 for lanes 0-15, 2 for lanes 16-31
    const int rowA  = 16 * wv + lhalf;   // batch row this lane feeds into A

    // ----- layer 1: K=256 in steps of 4 -----
    v8f acc_a = {}; v8f acc_m = {};
    for (int k0 = 0; k0 < CCH; k0 += 4) {
        v2f a_av, a_mx, bf;
        a_av.x = avg[rowA * CCH + k0 + khi];
        a_av.y = avg[rowA * CCH + k0 + khi + 1];
        a_mx.x = mxp[rowA * CCH + k0 + khi];
        a_mx.y = mxp[rowA * CCH + k0 + khi + 1];
        // B[k][n] = w1[n][k]
        bf.x = w1[lhalf * CCH + k0 + khi];
        bf.y = w1[lhalf * CCH + k0 + khi + 1];
        acc_a = __builtin_amdgcn_wmma_f32_16x16x4_f32(false, a_av, false, bf,
                                                      (short)0, acc_a, false, false);
        acc_m = __builtin_amdgcn_wmma_f32_16x16x4_f32(false, a_mx, false, bf,
                                                      (short)0, acc_m, false, false);
    }
    // relu + C/D-layout store to LDS (VGPR r -> row r or r+8, split at lane 16)
    const int mbase = 16 * wv + ((lane >> 4) ? 8 : 0);
#pragma unroll
    for (int r = 0; r < 8; ++r) {
        float ha = acc_a[r]; ha = ha > 0.0f ? ha : 0.0f;
        float hm = acc_m[r]; hm = hm > 0.0f ? hm : 0.0f;
        h_lds[mbase + r][lhalf]      = ha;   // avg block rows 0..31
        h_lds[32 + mbase + r][lhalf] = hm;   // max block rows 32..63
    }
    __syncthreads();

    // ----- layer 2: 16 N-tiles, K=16 in steps of 4 -----
    for (int nt = 0; nt < 16; ++nt) {
        const int n0 = nt * 16;
        v8f oa = {}; v8f om = {};
#pragma unroll
        for (int ks = 0; ks < 4; ++ks) {
            const int k0 = ks * 4;
            v2f a_av, a_mx, bf;
            a_av.x = h_lds[rowA][k0 + khi];
            a_av.y = h_lds[rowA][k0 + khi + 1];
            a_mx.x = h_lds[32 + rowA][k0 + khi];
            a_mx.y = h_lds[32 + rowA][k0 + khi + 1];
            // B[k][n] = w2[n0+n][k]
            bf.x = w2[(n0 + lhalf) * 16 + k0 + khi];
            bf.y = w2[(n0 + lhalf) * 16 + k0 + khi + 1];
            oa = __builtin_amdgcn_wmma_f32_16x16x4_f32(false, a_av, false, bf,
                                                       (short)0, oa, false, false);
            om = __builtin_amdgcn_wmma_f32_16x16x4_f32(false, a_mx, false, bf,
                                                       (short)0, om, false, false);
        }
#pragma unroll
        for (int r = 0; r < 8; ++r) {
            float s = oa[r] + om[r];
            float v = 1.0f / (1.0f + __expf(-s));
            ca[(mbase + r) * CCH + n0 + lhalf] = v;   // batch rows 0..31
        }
    }
}

// -------- Kernel 3a: spatial pooled = {mean_c, max_c} of x*ca ---------------
__global__ __launch_bounds__(256) void cbam_sppool(const float* __restrict__ x,
                                                   const float* __restrict__ ca,
                                                   float* __restrict__ pavg,
                                                   float* __restrict__ pmax) {
    __shared__ float s_ca[CCH];
    const int b = blockIdx.x / NTILE;
    const int t = blockIdx.x % NTILE;
    s_ca[threadIdx.x] = ca[b * CCH + threadIdx.x];
    __syncthreads();
    const int hw = t * 256 + threadIdx.x;
    if (hw >= HW) return;
    const float* px = x + (size_t)b * CCH * HW + hw;
    float sum = 0.0f, mx = -3.402823466e38f;
#pragma unroll 8
    for (int c = 0; c < CCH; ++c) {
        float v = px[(size_t)c * HW] * s_ca[c];     // coalesced across the wave
        sum += v;
        mx = fmaxf(mx, v);
    }
    pavg[b * HW + hw] = sum * (1.0f / (float)CCH);
    pmax[b * HW + hw] = mx;
}

// -------- Kernel 3b: 7x7 conv (2->1 ch) + sigmoid -> sa ---------------------
__global__ __launch_bounds__(256) void cbam_conv_sa(const float* __restrict__ pavg,
                                                    const float* __restrict__ pmax,
                                                    const float* __restrict__ wsp, // [1,2,7,7]
                                                    float* __restrict__ sa) {
    __shared__ float s_w[98];
    if (threadIdx.x < 98) s_w[threadIdx.x] = wsp[threadIdx.x];
    __syncthreads();
    const int b = blockIdx.x / NTILE;
    const int t = blockIdx.x % NTILE;
    const int hw = t * 256 + threadIdx.x;
    if (hw >= HW) return;
    const int h = hw / 56, w = hw % 56;
    const float* pa = pavg + b * HW;
    const float* pm = pmax + b * HW;
    float acc = 0.0f;
#pragma unroll
    for (int kh = 0; kh < 7; ++kh) {
        const int ih = h + kh - 3;
        if (ih < 0 || ih >= 56) continue;
#pragma unroll
        for (int kw = 0; kw < 7; ++kw) {
            const int iw = w + kw - 3;
            if (iw < 0 || iw >= 56) continue;
            const int idx = ih * 56 + iw;
            acc += s_w[kh * 7 + kw] * pa[idx] + s_w[49 + kh * 7 + kw] * pm[idx];
        }
    }
    sa[b * HW + hw] = 1.0f / (1.0f + __expf(-acc));
}

// -------- Kernel 3c: out = x * (1 + ca*sa)  (float4 streaming) --------------
__global__ __launch_bounds__(256) void cbam_final(const float* __restrict__ x,
                                                  const float* __restrict__ ca,
                                                  const float* __restrict__ sa,
                                                  float* __restrict__ out) {
    const int idx4 = blockIdx.x * 256 + threadIdx.x;     // 0 .. B*C*HW/4 - 1
    const int hw4 = idx4 % HW4;
    const int bc  = idx4 / HW4;
    const int b   = bc >> 8;
    const float4 x4  = ((const float4*)x)[idx4];
    const float  cav = ca[bc];
    const float4 s4  = ((const float4*)(sa + b * HW))[hw4];
    float4 o;
    o.x = x4.x * (1.0f + cav * s4.x);
    o.y = x4.y * (1.0f + cav * s4.y);
    o.z = x4.z * (1.0f + cav * s4.z);
    o.w = x4.w * (1.0f + cav * s4.w);
    ((float4*)out)[idx4] = o;
}

// ---------------------------------------------------------------------------
extern "C" void kernel_launch(void* const* d_in, const int* in_sizes, int n_in,
                              void* d_out, int out_size, void* d_ws, size_t ws_size,
                              hipStream_t stream) {
    const float* x   = (const float*)d_in[0];   // (32,256,56,56)
    const float* w1  = (const float*)d_in[1];   // (16,256)
    const float* w2  = (const float*)d_in[2];   // (256,16)
    const float* wsp = (const float*)d_in[3];   // (1,2,7,7)
    float* out = (float*)d_out;

    float* ws   = (float*)d_ws;
    float* avg  = ws;                        // 8192
    float* mxp  = ws + 8192;                 // 8192
    float* ca   = ws + 16384;                // 8192
    float* pavg = ws + 24576;                // 100352
    float* pmax = ws + 24576 + 100352;       // 100352
    float* sa   = ws + 24576 + 200704;       // 100352  (total ~1.27 MB)

    cbam_chpool <<<32 * 256,   256, 0, stream>>>(x, avg, mxp);
    cbam_mlp_wmma<<<1,          64, 0, stream>>>(avg, mxp, w1, w2, ca);
    cbam_sppool <<<32 * NTILE, 256, 0, stream>>>(x, ca, pavg, pmax);
    cbam_conv_sa<<<32 * NTILE, 256, 0, stream>>>(pavg, pmax, wsp, sa);
    cbam_final  <<<25088,      256, 0, stream>>>(x, ca, sa, out);
}